// DTCWTForward_44126493999648
// MI455X (gfx1250) — compile-verified
//
#include <hip/hip_runtime.h>

typedef __attribute__((ext_vector_type(2))) float v2f;
typedef __attribute__((ext_vector_type(8))) float v8f;

#define INV_SQRT2f 0.70710678118654752440f

#if __has_builtin(__builtin_amdgcn_global_load_async_to_lds_b32) && \
    __has_builtin(__builtin_amdgcn_s_wait_asynccnt)
#define USE_ASYNC_LDS 1
typedef __attribute__((address_space(1))) int* gint_t;
typedef __attribute__((address_space(3))) int* lint_t;
#else
#define USE_ASYNC_LDS 0
#endif

// ---------------------------------------------------------------------------
// Level 0: phi = conv(x, h0o, pad 2,2), hi = conv(x, h1o, pad 3,3).
// psi0 (complex64) is memory-identical to hi, so hi goes straight to d_out.
// ---------------------------------------------------------------------------
__global__ __launch_bounds__(256) void dtcwt_level0(
    const float* __restrict__ x, const float* __restrict__ h0o,
    const float* __restrict__ h1o, float* __restrict__ phi0,
    float* __restrict__ psi0, int T)
{
    const int r  = blockIdx.y;
    const int t0 = (blockIdx.x * 256 + threadIdx.x) * 4;
    const float* xr = x + (size_t)r * T;

    float h0[5], h1[7];
#pragma unroll
    for (int k = 0; k < 5; ++k) h0[k] = h0o[k];
#pragma unroll
    for (int k = 0; k < 7; ++k) h1[k] = h1o[k];

    float w[10];                                   // x[t0-3 .. t0+6]
#pragma unroll
    for (int j = 0; j < 10; ++j) {
        int g = t0 - 3 + j;
        w[j] = ((unsigned)g < (unsigned)T) ? xr[g] : 0.0f;
    }

    float lo[4], hi[4];
#pragma unroll
    for (int i = 0; i < 4; ++i) {
        float a = 0.0f, b = 0.0f;
#pragma unroll
        for (int k = 0; k < 5; ++k) a = fmaf(h0[k], w[i + k + 1], a); // x[t+k-2]
#pragma unroll
        for (int k = 0; k < 7; ++k) b = fmaf(h1[k], w[i + k], b);     // x[t+k-3]
        lo[i] = a; hi[i] = b;
    }
    *(float4*)(phi0 + (size_t)r * T + t0) = make_float4(lo[0], lo[1], lo[2], lo[3]);
    *(float4*)(psi0 + (size_t)r * T + t0) = make_float4(hi[0], hi[1], hi[2], hi[3]);
}

// ---------------------------------------------------------------------------
// Levels 1..7: four stride-2 10-tap convs as banded-Toeplitz WMMA matmuls.
//   waves 0-3 : a-branch (even phi samples), position tiles 0..3 (16 pos each)
//   waves 4-7 : b-branch (odd  phi samples), same tiles
// Output is transposed through LDS (aliasing the input tile) so global
// stores are coalesced float4.
// ---------------------------------------------------------------------------
#define TILE_W 272
#define TILE_S 277     // input tile LDS stride (277 mod 64 = 21, odd)
#define OUT_S  132     // output tile LDS stride (16B-aligned rows)

__global__ __launch_bounds__(256) void dtcwt_level(
    const float* __restrict__ phi_in, float* __restrict__ phi_out,
    float* __restrict__ psi_out,
    const float* __restrict__ f0a, const float* __restrict__ f0b,
    const float* __restrict__ f1a, const float* __restrict__ f1b,
    int Lin, float conjSign)
{
    __shared__ float smem[16 * TILE_S];    // phase A: input tile; phase B: outputs
    __shared__ float sF[4][16];            // f0a,f0b,f1a,f1b zero-padded to 16

    float* tile = smem;
    float* oPhi = smem;                    // 16 x OUT_S
    float* oPsi = smem + 16 * OUT_S;       // 16 x OUT_S  (2*16*132 <= 16*277)

    const int tid     = threadIdx.x;
    const int rowBase = blockIdx.y * 16;
    const int P0      = blockIdx.x * 64;   // first output position of this block

    if (tid < 64) {
        int f = tid >> 4, k = tid & 15;
        const float* src = (f == 0) ? f0a : (f == 1) ? f0b : (f == 2) ? f1a : f1b;
        sF[f][k] = (k < 10) ? src[k] : 0.0f;
    }

    // --- Phase A: stage 16 rows x 272 cols of phi, indices [4*P0-10, 4*P0+261]
    const int gBase = 4 * P0 - 10;
    for (int rr = 0; rr < 16; ++rr) {
        const float* rp = phi_in + (size_t)(rowBase + rr) * Lin;
        for (int col = tid; col < TILE_W; col += 256) {
            int g = gBase + col;
            float* lp = &tile[rr * TILE_S + col];
            if ((unsigned)g < (unsigned)Lin) {
#if USE_ASYNC_LDS
                __builtin_amdgcn_global_load_async_to_lds_b32(
                    (gint_t)(rp + g), (lint_t)lp, 0, 0);
#else
                *lp = rp[g];
#endif
            } else {
                *lp = 0.0f;                        // halo zero-fill
            }
            __builtin_prefetch(rp + g + 256, 0, 1); // global_prefetch_b8
        }
    }
#if USE_ASYNC_LDS
    __builtin_amdgcn_s_wait_asynccnt(0);
#endif
    __syncthreads();

    const int wave  = tid >> 5;
    const int lane  = tid & 31;
    const int wt    = wave & 3;             // position tile within block
    const int bb    = wave >> 2;            // 0 = a-branch, 1 = b-branch
    const int p     = lane & 15;            // A row (pos) / B,D col (channel)
    const int khalf = (lane >> 4) << 1;     // K offset: 0 (lanes 0-15) / 2 (16-31)

    const float* F0 = bb ? sF[1] : sF[0];
    const float* F1 = bb ? sF[3] : sF[2];

    // Banded-Toeplitz A chunks: A[p, 4c+k] = h[4c+k - 2p]
    v2f A0[10], A1[10];
#pragma unroll
    for (int c = 0; c < 10; ++c) {
        int i0 = 4 * c + khalf - 2 * p;
        int i1 = i0 + 1;
        float a00 = F0[i0 & 15]; a00 = ((unsigned)i0 < 10u) ? a00 : 0.0f;
        float a01 = F0[i1 & 15]; a01 = ((unsigned)i1 < 10u) ? a01 : 0.0f;
        float a10 = F1[i0 & 15]; a10 = ((unsigned)i0 < 10u) ? a10 : 0.0f;
        float a11 = F1[i1 & 15]; a11 = ((unsigned)i1 < 10u) ? a11 : 0.0f;
        A0[c].x = a00; A0[c].y = a01;
        A1[c].x = a10; A1[c].y = a11;
    }

    // K-loop: B[c][k, n] = tile[n][64*wt + 8c + 2*(khalf+k) + bb]
    v8f dlo = {};
    v8f dhi = {};
    const int nrow    = p * TILE_S;
    const int colBase = 64 * wt + 2 * khalf + bb;
#pragma unroll
    for (int c = 0; c < 10; ++c) {
        v2f b;
        int col = colBase + 8 * c;
        b.x = tile[nrow + col];
        b.y = tile[nrow + col + 2];
        dlo = __builtin_amdgcn_wmma_f32_16x16x4_f32(
                  false, A0[c], false, b, (short)0, dlo, false, false);
        dhi = __builtin_amdgcn_wmma_f32_16x16x4_f32(
                  false, A1[c], false, b, (short)0, dhi, false, false);
    }

    __syncthreads();   // all tile reads done; smem now reused for outputs

    // --- Phase B: D -> LDS (interleaved a/b), rows = channels, 128 cols
    const float sLo = INV_SQRT2f;
    const float sHi = INV_SQRT2f * (bb ? conjSign : 1.0f);
    const int   cl0 = 32 * wt + ((lane >> 4) << 4) + bb;  // local col of (v=0)
    const int   orow = p * OUT_S;
#pragma unroll
    for (int v = 0; v < 8; ++v) {
        oPhi[orow + cl0 + 2 * v] = dlo[v] * sLo;
        oPsi[orow + cl0 + 2 * v] = dhi[v] * sHi;
    }
    __syncthreads();

    // --- Coalesced float4 stores: 16 rows x 128 floats per array
    const int L2 = Lin >> 1;
    for (int idx = tid; idx < 1024; idx += 256) {
        int arr = idx >> 9;                 // 0 = phi, 1 = psi
        int sub = idx & 511;
        int row = sub >> 5;
        int q   = sub & 31;
        const float* src = (arr ? oPsi : oPhi) + row * OUT_S + 4 * q;
        float4 v4 = *(const float4*)src;
        float* dst = (arr ? psi_out : phi_out)
                   + (size_t)(rowBase + row) * L2 + 2 * P0 + 4 * q;
        *(float4*)dst = v4;
    }
}

// ---------------------------------------------------------------------------
extern "C" void kernel_launch(void* const* d_in, const int* in_sizes, int n_in,
                              void* d_out, int out_size, void* d_ws, size_t ws_size,
                              hipStream_t stream)
{
    const int T = 131072;
    const int R = in_sizes[0] / T;   // 256 rows (B*C)

    const float* x   = (const float*)d_in[0];
    const float* h0o = (const float*)d_in[1];
    const float* h1o = (const float*)d_in[2];
    const float* h0a = (const float*)d_in[3];
    const float* h0b = (const float*)d_in[4];
    const float* h1a = (const float*)d_in[5];
    const float* h1b = (const float*)d_in[6];
    const float* g0a = (const float*)d_in[7];
    const float* g0b = (const float*)d_in[8];
    const float* g1a = (const float*)d_in[9];
    const float* g1b = (const float*)d_in[10];

    float* out  = (float*)d_out;
    float* phiA = (float*)d_ws;                   // R*T floats   (128 MB)
    float* phiB = phiA + (size_t)R * T;           // R*T/2 floats ( 64 MB)

    // Output layout (floats): phi | psi0 | psi1 | ... | psi7
    size_t psiOff[8];
    psiOff[0] = (size_t)R * (T >> 7);             // after final phi
    size_t cur = psiOff[0] + (size_t)R * T;       // psi0 = R*T floats
    for (int j = 1; j < 8; ++j) { psiOff[j] = cur; cur += (size_t)R * (T >> j); }

    // Level 0: phi -> ws, hi -> psi0 region directly.
    dtcwt_level0<<<dim3(T / 1024, R), 256, 0, stream>>>(
        x, h0o, h1o, phiA, out + psiOff[0], T);

    // Levels 1..7, ping-pong phi between phiA/phiB; final phi -> d_out.
    for (int j = 1; j < 8; ++j) {
        int Lin  = T >> (j - 1);
        int Lout = Lin >> 2;
        const float* pin  = (j & 1) ? phiA : phiB;
        float*       pout = (j == 7) ? out : ((j & 1) ? phiB : phiA);
        const float* F0a  = (j & 1) ? g0a : h0a;
        const float* F0b  = (j & 1) ? g0b : h0b;
        const float* F1a  = (j & 1) ? g1a : h1a;
        const float* F1b  = (j & 1) ? g1b : h1b;
        float cs = (j & 1) ? -1.0f : 1.0f;        // conj on odd levels
        dtcwt_level<<<dim3(Lout / 64, R / 16), 256, 0, stream>>>(
            pin, pout, out + psiOff[j], F0a, F0b, F1a, F1b, Lin, cs);
    }
}